// LogEig_65317862638298
// MI455X (gfx1250) — compile-verified
//
#include <hip/hip_runtime.h>
#include <stdint.h>

#define DIM 64
#define S   68          // LDS row stride in floats (64 + 4 pad -> no bank conflicts)
#define NTHREADS 256    // 8 wave32 per block

typedef float v2f __attribute__((ext_vector_type(2)));
typedef float v8f __attribute__((ext_vector_type(8)));

// C(16x16) tiles: wave w owns tiles 2w and 2w+1 of the 4x4 tile grid (same tile
// row -> A fragments shared). acc[tt] = A x B over K=64 via V_WMMA_F32_16X16X4_F32.
__device__ __forceinline__ void mm64_acc(const float* __restrict__ A,
                                         const float* __restrict__ Bm,
                                         v8f acc[2])
{
#if defined(__gfx1250__)
    const int lane = threadIdx.x & 31;
    const int wave = threadIdx.x >> 5;
    const int m15  = lane & 15;
    const int hi   = lane >> 4;
    #pragma unroll
    for (int tt = 0; tt < 2; ++tt) {
        const int t  = wave * 2 + tt;
        const int tr = (t >> 2) << 4;
        const int tc = (t & 3) << 4;
        const int am = tr + m15;          // A row this lane supplies
        v8f c = {0.f, 0.f, 0.f, 0.f, 0.f, 0.f, 0.f, 0.f};
        #pragma unroll
        for (int kt = 0; kt < 16; ++kt) {
            const int k0 = kt * 4 + 2 * hi;   // ISA 32-bit A 16x4 layout
            v2f a, b;
            a.x = A[am * S + k0];
            a.y = A[am * S + k0 + 1];
            b.x = Bm[(k0    ) * S + tc + m15];
            b.y = Bm[(k0 + 1) * S + tc + m15];
            c = __builtin_amdgcn_wmma_f32_16x16x4_f32(
                    false, a, false, b, (short)0, c, false, false);
        }
        acc[tt] = c;
    }
#else
    (void)A; (void)Bm;
    acc[0] = (v8f){0,0,0,0,0,0,0,0}; acc[1] = acc[0];
#endif
}

// C = scale*acc + idAdd*I  (scale/idAdd fold per inlined call site)
__device__ __forceinline__ void store_tiles(float* __restrict__ C,
                                            const v8f acc[2],
                                            float scale, float idAdd)
{
    const int lane = threadIdx.x & 31;
    const int wave = threadIdx.x >> 5;
    const int m15  = lane & 15;
    const int hi   = lane >> 4;
    #pragma unroll
    for (int tt = 0; tt < 2; ++tt) {
        const int t  = wave * 2 + tt;
        const int tr = (t >> 2) << 4;
        const int tc = (t & 3) << 4;
        #pragma unroll
        for (int v = 0; v < 8; ++v) {
            const int m = tr + v + 8 * hi;    // ISA 32-bit C/D 16x16 layout
            const int n = tc + m15;
            float val = scale * acc[tt][v];
            if (m == n) val += idAdd;
            C[m * S + n] = val;
        }
    }
}

__global__ __launch_bounds__(NTHREADS)
void logeig_kernel(const float* __restrict__ X, float* __restrict__ out, int nb)
{
    __shared__ float Ybuf[DIM * S];
    __shared__ float Zbuf[DIM * S];
    __shared__ float Wbuf[DIM * S];
    const int b   = blockIdx.x;
    const int tid = threadIdx.x;
    if (b >= nb) return;

    // ---- Load one 64x64 f32 matrix into Ybuf (row stride 68) via TDM ----
#if defined(__gfx1250__) && __has_builtin(__builtin_amdgcn_tensor_load_to_lds) && __has_builtin(__builtin_amdgcn_s_wait_tensorcnt)
    if (tid < 32) {   // one wave issues the TDM descriptor
        typedef unsigned int u32x4 __attribute__((ext_vector_type(4)));
        typedef int          i32x4 __attribute__((ext_vector_type(4)));
        typedef int          i32x8 __attribute__((ext_vector_type(8)));
        const unsigned long long ga =
            (unsigned long long)(uintptr_t)(X + (size_t)b * (DIM * DIM));
        const unsigned int ldsa = (unsigned int)(uintptr_t)&Ybuf[0];
        u32x4 g0;
        g0[0] = 1u;                                            // count=1 (valid D#)
        g0[1] = ldsa;                                          // lds_addr
        g0[2] = (unsigned int)(ga & 0xFFFFFFFFull);            // global_addr[31:0]
        g0[3] = (unsigned int)((ga >> 32) & 0x1FFFFFFull)      // global_addr[56:32]
              | (2u << 30);                                    // type=2 ("image")
        i32x8 g1;
        g1[0] = (2 << 16)        // data_size = 4B
              | (1 << 20)        // pad_enable
              | (5 << 22)        // pad_interval: every 64 DWORDs (one row)
              | (3 << 25);       // pad_amount: 4 DWORDs -> LDS stride 68 floats
        g1[1] = (int)(64u << 16);  // tensor_dim0 = 64
        g1[2] = (int)(64u << 16);  // tensor_dim1 = 64
        g1[3] = (int)(64u << 16);  // tile_dim0   = 64
        g1[4] = 64;                // tile_dim1 = 64, tile_dim2 = 0
        g1[5] = 64;                // tensor_dim0_stride = 64 elements
        g1[6] = 0;
        g1[7] = 0;
        i32x4 gz4 = {0, 0, 0, 0};          // groups 2/3 unused (2-D tensor)
        i32x8 gz8 = {0, 0, 0, 0, 0, 0, 0, 0};
        __builtin_amdgcn_tensor_load_to_lds(g0, g1, gz4, gz4, gz8, 0);
        __builtin_amdgcn_s_wait_tensorcnt(0);
    }
#else
    for (int i = tid; i < DIM * DIM; i += NTHREADS)
        Ybuf[(i >> 6) * S + (i & 63)] = X[(size_t)b * (DIM * DIM) + i];
#endif
    __syncthreads();

    // Y = X/6 (spectrum in (0,1)) ; Z = I
    for (int i = tid; i < DIM * DIM; i += NTHREADS) {
        const int r = i >> 6, c = i & 63;
        Ybuf[r * S + c] *= (1.0f / 6.0f);
        Zbuf[r * S + c] = (r == c) ? 1.0f : 0.0f;
    }
    __syncthreads();

    // ---- 4 stages of coupled Newton-Schulz sqrt: Y -> (X/6)^(1/16) ----
    const int ns_iters[4] = {7, 5, 4, 3};
    v8f accW[2], accY[2], accZ[2];
    #pragma clang loop unroll(disable)
    for (int stage = 0; stage < 4; ++stage) {
        if (stage > 0) {                       // restart with Z = I
            for (int i = tid; i < DIM * DIM; i += NTHREADS) {
                const int r = i >> 6, c = i & 63;
                Zbuf[r * S + c] = (r == c) ? 1.0f : 0.0f;
            }
            __syncthreads();
        }
        const int ni = ns_iters[stage];
        #pragma clang loop unroll(disable)
        for (int it = 0; it < ni; ++it) {
            mm64_acc(Zbuf, Ybuf, accW);                // Z*Y
            store_tiles(Wbuf, accW, -0.5f, 1.5f);      // W = 1.5I - 0.5*Z*Y
            __syncthreads();
            mm64_acc(Ybuf, Wbuf, accY);                // Ynew = Y*W  (regs)
            mm64_acc(Wbuf, Zbuf, accZ);                // Znew = W*Z  (regs)
            __syncthreads();                           // all reads of Y,Z done
            store_tiles(Ybuf, accY, 1.0f, 0.0f);
            store_tiles(Zbuf, accZ, 1.0f, 0.0f);
            __syncthreads();
        }
    }

    // ---- log(Y) via Mercator series, E = Y - I, |spec(E)| <= ~0.11 ----
    // Zbuf <- E (Z no longer needed), Wbuf <- Q1 = 0.2E - 0.25I
    for (int i = tid; i < DIM * DIM; i += NTHREADS) {
        const int r = i >> 6, c = i & 63;
        const float e = Ybuf[r * S + c] - ((r == c) ? 1.0f : 0.0f);
        Zbuf[r * S + c] = e;
        Wbuf[r * S + c] = 0.2f * e - ((r == c) ? 0.25f : 0.0f);
    }
    __syncthreads();
    // Q <- E*Q + c_k*I (Horner): c = 1/3, -1/2, 1, 0 ; final L = E*Q in Wbuf
    const float idc[4] = { 1.0f / 3.0f, -0.5f, 1.0f, 0.0f };
    #pragma clang loop unroll(disable)
    for (int h = 0; h < 4; ++h) {
        mm64_acc(Zbuf, Wbuf, accW);                    // E * Q
        __syncthreads();
        store_tiles(Wbuf, accW, 1.0f, idc[h]);
        __syncthreads();
    }

    // ---- logX = 16*W + log(6)*I ; emit diag then tril(k=-1)*sqrt(2) ----
    const size_t ob = (size_t)b * 2080;
    for (int i = tid; i < DIM; i += NTHREADS)
        out[ob + i] = 16.0f * Wbuf[i * S + i] + 1.7917594692280550f; // +log 6
    for (int j = tid; j < 2016; j += NTHREADS) {
        int r = (int)((1.0f + __builtin_sqrtf(1.0f + 8.0f * (float)j)) * 0.5f);
        while (r * (r - 1) / 2 > j) --r;
        while ((r + 1) * r / 2 <= j) ++r;
        const int c = j - r * (r - 1) / 2;
        out[ob + DIM + j] = 22.627416997969522f * Wbuf[r * S + c];   // 16*sqrt(2)
    }
}

extern "C" void kernel_launch(void* const* d_in, const int* in_sizes, int n_in,
                              void* d_out, int out_size, void* d_ws, size_t ws_size,
                              hipStream_t stream)
{
    (void)n_in; (void)out_size; (void)d_ws; (void)ws_size;
    const float* X = (const float*)d_in[0];
    float* out = (float*)d_out;
    const int nb = in_sizes[0] / (DIM * DIM);   // 16384 batches
    logeig_kernel<<<nb, NTHREADS, 0, stream>>>(X, out, nb);
}